// LUTGEMMLinear_50165218017639
// MI455X (gfx1250) — compile-verified
//
#include <hip/hip_runtime.h>
#include <stdint.h>

#define IN_F      4096
#define OUT_F     11008
#define BITS      4
#define GROUPSZ   128
#define N_WORDS   (IN_F / 32)          // 128
#define N_GROUPS  (IN_F / GROUPSZ)     // 32

#define TILE_O         256
#define WORDS_PER_BLK  16              // 4 full groups of 128 inputs
#define GROUPS_PER_BLK 4
#define ROWS_PER_BLK   (WORDS_PER_BLK * BITS)   // 64 qweight rows per block
#define GZ             (N_WORDS / WORDS_PER_BLK) // 8 word-chunks

// LDS layout (byte offsets into dynamic shared segment)
#define XS_OFF    0                         // 512 floats  = 2048 B
#define GSUM_OFF  2048                      // 4 floats    = 16 B
#define LUT_OFF   2064                      // 16384 floats= 65536 B
#define QT_OFF    (LUT_OFF + 65536)         // 67600, 16B-aligned; 64*256 ints
#define SMEM_BYTES (QT_OFF + 65536)         // 133136 B (fits 320KB WGP LDS)

typedef unsigned int v4u __attribute__((ext_vector_type(4)));
typedef int          v8i __attribute__((ext_vector_type(8)));
typedef int          v4i __attribute__((ext_vector_type(4)));

__global__ void zero_f32_kernel(float* __restrict__ p, int n) {
    int i = blockIdx.x * blockDim.x + threadIdx.x;
    if (i < n) p[i] = 0.0f;
}

__global__ __launch_bounds__(256)
void lutgemm_bcq_kernel(const float* __restrict__ x,
                        const int*   __restrict__ qweight,
                        const float* __restrict__ alpha,
                        const float* __restrict__ qbias,
                        float*       __restrict__ out) {
    extern __shared__ char smem[];
    float* xs   = (float*)(smem + XS_OFF);    // x slice: 512 floats
    float* gsum = (float*)(smem + GSUM_OFF);  // per-group sums: 4 floats
    float* lut  = (float*)(smem + LUT_OFF);   // 16 words * 4 chunks * 256 entries
    int*   qt   = (int*)  (smem + QT_OFF);    // 64 rows * 256 cols int32

    const int tid = threadIdx.x;
    const int ox  = blockIdx.x;   // output tile index (0..42)
    const int gz  = blockIdx.y;   // word-chunk index  (0..7)
    const int o   = ox * TILE_O + tid;

    // ---- Wave 0: launch TDM load of 64x256 int32 qweight tile into LDS ----
    // 2D descriptor: tensor row length OUT_F (elements), row stride OUT_F,
    // tile = 256 cols x 64 rows, data_size = 4 bytes, tile start = row gz*64,
    // col ox*256. Tracked by TENSORcnt; overlaps with LUT construction below.
    if (tid < 32) {
        uint64_t ga = (uint64_t)(uintptr_t)qweight +
                      (((uint64_t)gz * ROWS_PER_BLK) * OUT_F +
                       (uint64_t)ox * TILE_O) * 4ull;
        unsigned lds_addr = (unsigned)(uintptr_t)qt;  // low 32 bits = LDS offset

        v4u g0;
        g0[0] = 1u;                                   // count=1, user descriptor
        g0[1] = lds_addr;                             // lds_addr
        g0[2] = (unsigned)(ga & 0xFFFFFFFFu);         // global_addr[31:0]
        g0[3] = (unsigned)((ga >> 32) & 0x01FFFFFFu)  // global_addr[56:32]
              | (2u << 30);                           // type=2 ("image")

        v8i g1;
        g1[0] = (int)(2u << 16);                      // wg_mask=0, data_size=2 (4B)
        g1[1] = (int)(((unsigned)OUT_F & 0xFFFFu) << 16);          // tensor_dim0 lo16
        g1[2] = (int)(((unsigned)OUT_F >> 16) |
                      ((unsigned)(N_WORDS * BITS) << 16));         // dim0 hi | tensor_dim1 lo16 (512)
        g1[3] = (int)((unsigned)TILE_O << 16);        // tile_dim0 = 256
        g1[4] = ROWS_PER_BLK;                         // tile_dim1 = 64, tile_dim2 = 0
        g1[5] = OUT_F;                                // tensor_dim0_stride[31:0]
        g1[6] = 0;                                    // stride hi | dim1_stride lo (unused, 2D)
        g1[7] = 0;
        v4i g2 = {0, 0, 0, 0};                        // 2D: dims 2/3 unused
        v4i g3 = {0, 0, 0, 0};
        v8i g4 = {0, 0, 0, 0, 0, 0, 0, 0};            // trailing group (clang-23 6-arg form)
        __builtin_amdgcn_tensor_load_to_lds(g0, g1, g2, g3, g4, /*cpol=*/0);
    }

    // ---- Stage the 512-float x slice for this word-chunk into LDS ----
    const float* xblk = x + (size_t)gz * (WORDS_PER_BLK * 32);
    xs[tid]       = xblk[tid];
    xs[tid + 256] = xblk[tid + 256];
    __syncthreads();

    // ---- Build the 8-bit LUT while the TDM transfer is in flight ----
    // tid -> (pair = word*4+chunk in [0,64), quarter in [0,4)); each thread
    // produces 64 contiguous entries from 8 registered x values.
    {
        const int pair    = tid >> 2;
        const int quarter = tid & 3;
        const float* xv = xs + (pair << 3);
        const float x0 = xv[0], x1 = xv[1], x2 = xv[2], x3 = xv[3];
        const float x4 = xv[4], x5 = xv[5], x6 = xv[6], x7 = xv[7];
        float base = ((quarter & 1) ? x6 : 0.0f) + ((quarter & 2) ? x7 : 0.0f);
        float* dst = lut + pair * 256 + quarter * 64;
#pragma unroll 8
        for (int k = 0; k < 64; ++k) {
            float s = base;
            s += (k & 1)  ? x0 : 0.0f;
            s += (k & 2)  ? x1 : 0.0f;
            s += (k & 4)  ? x2 : 0.0f;
            s += (k & 8)  ? x3 : 0.0f;
            s += (k & 16) ? x4 : 0.0f;
            s += (k & 32) ? x5 : 0.0f;
            dst[k] = s;
        }
    }

    // ---- Per-group x sums (4 groups of 128) ----
    if (tid < 4) {
        float s = 0.0f;
        for (int i = 0; i < GROUPSZ; ++i) s += xs[tid * GROUPSZ + i];
        gsum[tid] = s;
    }

    // Issuing wave drains the tensor DMA, then whole block syncs.
    if (tid < 32) __builtin_amdgcn_s_wait_tensorcnt(0);
    __syncthreads();

    // ---- Main accumulation: 1 output column per thread ----
    float acc = 0.0f;
    const float* aptr = alpha + ((size_t)gz * GROUPS_PER_BLK * BITS) * OUT_F + o;
    const float* bptr = qbias + ((size_t)gz * GROUPS_PER_BLK) * OUT_F + o;

#pragma unroll
    for (int g = 0; g < GROUPS_PER_BLK; ++g) {
        const float gs = gsum[g];
#pragma unroll
        for (int b = 0; b < BITS; ++b) {
            float p = 0.0f;
#pragma unroll
            for (int wl = 0; wl < 4; ++wl) {
                const int w = g * 4 + wl;                         // local word 0..15
                const unsigned q = (unsigned)qt[(w * BITS + b) * TILE_O + tid];
                const float* l = lut + w * 1024;                  // 4 chunks * 256
                p += l[        (q       & 255u)];
                p += l[256u + ((q >> 8)  & 255u)];
                p += l[512u + ((q >> 16) & 255u)];
                p += l[768u +  (q >> 24)        ];
            }
            // sum_i x_i*s_i over the group = 2*(set-bit sums) - group x-sum
            acc += aptr[(size_t)(g * BITS + b) * OUT_F] * (2.0f * p - gs);
        }
        acc += gs * bptr[(size_t)g * OUT_F];
    }

    atomicAdd(out + o, acc);
}

extern "C" void kernel_launch(void* const* d_in, const int* in_sizes, int n_in,
                              void* d_out, int out_size, void* d_ws, size_t ws_size,
                              hipStream_t stream) {
    const float* x      = (const float*)d_in[0];
    const int*   qw     = (const int*)  d_in[1];
    const float* alpha  = (const float*)d_in[2];
    const float* qbias  = (const float*)d_in[3];
    float*       out    = (float*)d_out;

    zero_f32_kernel<<<(OUT_F + 255) / 256, 256, 0, stream>>>(out, OUT_F);

    dim3 grid(OUT_F / TILE_O, GZ);   // 43 x 8 = 344 blocks
    lutgemm_bcq_kernel<<<grid, 256, SMEM_BYTES, stream>>>(x, qw, alpha, qbias, out);
}